// GCNEncoder_39573828666116
// MI455X (gfx1250) — compile-verified
//
#include <hip/hip_runtime.h>
#include <hip/hip_bf16.h>

typedef __attribute__((ext_vector_type(2))) float v2f;
typedef __attribute__((ext_vector_type(8))) float v8f;

// ---------------------------------------------------------------------------
// Guaranteed hardware fp32 atomic add (non-returning, STOREcnt-tracked).
// Avoids any CAS-loop fallback the generic atomicAdd lowering might pick.
// ---------------------------------------------------------------------------
__device__ __forceinline__ void atomic_add_f32_nr(float* addr, float val) {
    asm volatile("global_atomic_add_f32 %0, %1, off"
                 :: "v"(addr), "v"(val) : "memory");
}

// ---------------------------------------------------------------------------
// Degree / normalization kernels
// ---------------------------------------------------------------------------
__global__ void k_fill(float* p, int n, float v) {
    int i = blockIdx.x * blockDim.x + threadIdx.x;
    if (i < n) p[i] = v;
}

__global__ void k_degree(const long long* __restrict__ dst, float* __restrict__ deg, int E) {
    int e = blockIdx.x * blockDim.x + threadIdx.x;
    if (e < E) atomic_add_f32_nr(&deg[(int)dst[e]], 1.0f);
}

__global__ void k_rsqrt_inplace(float* __restrict__ deg, int n) {
    int i = blockIdx.x * blockDim.x + threadIdx.x;
    if (i < n) {
        float d = deg[i];                 // >= 1 always (self loop)
        deg[i] = (d > 0.0f) ? rsqrtf(d) : 0.0f;
    }
}

// ---------------------------------------------------------------------------
// fp32 WMMA GEMM: C[M x N] = act(A)[M x K] @ W[K x N]
// One wave32 per 16x16 tile of C; V_WMMA_F32_16X16X4_F32 along K, fully
// unrolled (K, N compile-time). Optional ReLU folded into the A-operand load.
// M multiple of 16. EXEC stays all-ones in active waves (wave-uniform exit).
// ---------------------------------------------------------------------------
template <int K, int N, bool RELU>
__global__ __launch_bounds__(256) void k_gemm_wmma_f32(
        const float* __restrict__ A, const float* __restrict__ W,
        float* __restrict__ C, int M) {
    const int waveId = (int)((blockIdx.x * blockDim.x + threadIdx.x) >> 5);
    const int lane   = (int)(threadIdx.x & 31);
    const int mtiles = M >> 4;
    constexpr int ntiles = N >> 4;            // power of two (8 or 4)
    if (waveId >= mtiles * ntiles) return;    // wave-uniform
    const int tm = waveId / ntiles;
    const int tn = waveId & (ntiles - 1);
    const int lo = lane & 15;                 // row (A) / col (B, C) in tile
    const int hi = lane >> 4;                 // K-half select (A/B), M-half (C/D)

    // A 16x4 f32 layout: VGPR0 -> K = k+2*hi, VGPR1 -> K = k+2*hi+1
    const float* arow = A + (size_t)(tm * 16 + lo) * K + 2 * hi;
    // B 4x16 f32 layout: rows striped across lanes, same K split
    const float* wcol = W + (size_t)(2 * hi) * N + tn * 16 + lo;

    v8f acc = {};
#pragma unroll
    for (int k = 0; k < K; k += 4) {
        float2 av = *(const float2*)(arow + k);       // one b64 load
        float a0 = av.x, a1 = av.y;
        if (RELU) { a0 = fmaxf(a0, 0.0f); a1 = fmaxf(a1, 0.0f); }
        v2f a; a[0] = a0; a[1] = a1;
        v2f b;
        b[0] = wcol[(size_t)k * N];                   // immediate offsets
        b[1] = wcol[(size_t)(k + 1) * N];
        // 8 args: (neg_a, A, neg_b, B, c_mod, C, reuse_a, reuse_b)
        acc = __builtin_amdgcn_wmma_f32_16x16x4_f32(
                false, a, false, b, (short)0, acc, false, false);
    }

    // C/D f32 layout: VGPR v -> row M = v + 8*hi, col N = lo
    float* crow = C + (size_t)(tm * 16 + 8 * hi) * N + tn * 16 + lo;
#pragma unroll
    for (int v = 0; v < 8; ++v) crow[v * N] = acc[v];
}

// ---------------------------------------------------------------------------
// out[i][c] = h[i][c] * dis[i]^2 + bias[c]   (self-loop term + bias; also
// initializes the output buffer before edge atomics). float4 per thread.
// LGC = log2(channels/4) so all index math is shifts/masks.
// ---------------------------------------------------------------------------
template <int LGC>
__global__ __launch_bounds__(256) void k_self_init(
        const float* __restrict__ h, const float* __restrict__ dis,
        const float* __restrict__ bias, float* __restrict__ out, int Nn) {
    constexpr int Cc = 4 << LGC;
    long long t = (long long)blockIdx.x * blockDim.x + threadIdx.x;
    int i = (int)(t >> LGC);
    if (i >= Nn) return;
    int c4 = ((int)t & ((1 << LGC) - 1)) << 2;
    float w = dis[i];
    float n2 = w * w;
    const float4 hv = *(const float4*)(h + (size_t)i * Cc + c4);
    const float4 bv = *(const float4*)(bias + c4);
    float4 ov;
    ov.x = fmaf(hv.x, n2, bv.x);
    ov.y = fmaf(hv.y, n2, bv.y);
    ov.z = fmaf(hv.z, n2, bv.z);
    ov.w = fmaf(hv.w, n2, bv.w);
    *(float4*)(out + (size_t)i * Cc + c4) = ov;
}

// ---------------------------------------------------------------------------
// Edge scatter: out[dst] += h[src] * dis[src]*dis[dst]
// One thread per (edge, 4-channel chunk): a wave covers one edge's full row,
// so the h[src] gather is a coalesced 512B burst; 4 HW fp32 atomics per lane.
// ---------------------------------------------------------------------------
template <int LGC>
__global__ __launch_bounds__(256) void k_edge_scatter(
        const float* __restrict__ h,
        const long long* __restrict__ src, const long long* __restrict__ dst,
        const float* __restrict__ dis, float* __restrict__ out, int E) {
    constexpr int Cc = 4 << LGC;
    long long t = (long long)blockIdx.x * blockDim.x + threadIdx.x;
    int e = (int)(t >> LGC);
    if (e >= E) return;
    int c4 = ((int)t & ((1 << LGC) - 1)) << 2;
    int s = (int)src[e];
    int d = (int)dst[e];
    float norm = dis[s] * dis[d];
    const float4 hv = *(const float4*)(h + (size_t)s * Cc + c4);  // b128 gather
    float* o = out + (size_t)d * Cc + c4;
    atomic_add_f32_nr(o + 0, hv.x * norm);
    atomic_add_f32_nr(o + 1, hv.y * norm);
    atomic_add_f32_nr(o + 2, hv.z * norm);
    atomic_add_f32_nr(o + 3, hv.w * norm);
}

// ---------------------------------------------------------------------------
// Launch
// ---------------------------------------------------------------------------
static inline int cdiv(long long a, long long b) { return (int)((a + b - 1) / b); }

extern "C" void kernel_launch(void* const* d_in, const int* in_sizes, int n_in,
                              void* d_out, int out_size, void* d_ws, size_t ws_size,
                              hipStream_t stream) {
    const float*     x    = (const float*)d_in[0];
    const long long* eidx = (const long long*)d_in[1];   // int64 [2, E]
    const float*     W1   = (const float*)d_in[2];
    const float*     b1   = (const float*)d_in[3];
    const float*     W2   = (const float*)d_in[4];
    const float*     b2   = (const float*)d_in[5];
    const float*     W3   = (const float*)d_in[6];
    const float*     b3   = (const float*)d_in[7];
    float*           out  = (float*)d_out;

    const int IN_C = 64, HID_C = 128, OUT_C = 64;
    const int N = in_sizes[0] / IN_C;        // 50000
    const int E = in_sizes[1] / 2;           // 800000
    const long long* src = eidx;
    const long long* dst = eidx + E;

    // workspace layout (all fully written before any read, every call)
    float* bufA = (float*)d_ws;                      // N * 128
    float* bufB = bufA + (size_t)N * HID_C;          // N * 128
    float* dis  = bufB + (size_t)N * HID_C;          // N

    const int B = 256;

    // ---- symmetric normalization (shared by all layers) ----
    k_fill<<<cdiv(N, B), B, 0, stream>>>(dis, N, 1.0f);           // self loop
    k_degree<<<cdiv(E, B), B, 0, stream>>>(dst, dis, E);
    k_rsqrt_inplace<<<cdiv(N, B), B, 0, stream>>>(dis, N);

    const long long gemmThreads128 = (long long)(N >> 4) * (128 >> 4) * 32;
    const long long gemmThreads64  = (long long)(N >> 4) * (64 >> 4) * 32;

    // ---- layer 1: x(64) -> 128 ----
    k_gemm_wmma_f32<64, 128, false><<<cdiv(gemmThreads128, B), B, 0, stream>>>(x, W1, bufA, N);
    k_self_init<5><<<cdiv((long long)N * 32, B), B, 0, stream>>>(bufA, dis, b1, bufB, N);
    k_edge_scatter<5><<<cdiv((long long)E * 32, B), B, 0, stream>>>(bufA, src, dst, dis, bufB, E);

    // ---- layer 2: relu(bufB)(128) -> 128 ----
    k_gemm_wmma_f32<128, 128, true><<<cdiv(gemmThreads128, B), B, 0, stream>>>(bufB, W2, bufA, N);
    k_self_init<5><<<cdiv((long long)N * 32, B), B, 0, stream>>>(bufA, dis, b2, bufB, N);
    k_edge_scatter<5><<<cdiv((long long)E * 32, B), B, 0, stream>>>(bufA, src, dst, dis, bufB, E);

    // ---- layer 3: relu(bufB)(128) -> 64, into d_out ----
    k_gemm_wmma_f32<128, 64, true><<<cdiv(gemmThreads64, B), B, 0, stream>>>(bufB, W3, bufA, N);
    k_self_init<4><<<cdiv((long long)N * 16, B), B, 0, stream>>>(bufA, dis, b3, out, N);
    k_edge_scatter<4><<<cdiv((long long)E * 16, B), B, 0, stream>>>(bufA, src, dst, dis, out, E);
}